// LimitedKVWithBBPM_29180007809705
// MI455X (gfx1250) — compile-verified
//
#include <hip/hip_runtime.h>
#include <hip/hip_bf16.h>
#include <stdint.h>

#define DIM        128
#define KV_CACHE   8192
#define BBPM_SIZE  131072
#define NTOK       (32 * 8192)   // 262144
#define BATCH      4             // rows per wave per async batch

// ---------------------------------------------------------------- async types / helpers
typedef int vi4 __attribute__((vector_size(16)));                 // 4 x i32 = 16B
typedef __attribute__((address_space(1))) vi4 gvi4;               // global
typedef __attribute__((address_space(3))) vi4 lvi4;               // LDS

#define CPOL_RT 0   // regular temporal
#define CPOL_NT 1   // TH_NT: non-temporal (gfx12+ CPol TH bits [2:0])

__device__ __forceinline__ void wait_async0() {
#if __has_builtin(__builtin_amdgcn_s_wait_asynccnt)
  __builtin_amdgcn_s_wait_asynccnt(0);
#else
  asm volatile("s_wait_asynccnt 0x0" ::: "memory");
#endif
}

// ---------------------------------------------------------------- init: mem = bbpm, table = -1
__global__ void bbpm_init_kernel(const float* __restrict__ bbpm,
                                 float* __restrict__ memF,
                                 int* __restrict__ table) {
  const size_t i = (size_t)blockIdx.x * blockDim.x + threadIdx.x; // grid == BBPM_SIZE*DIM
  memF[i] = bbpm[i];
  if (i < NTOK) table[i] = -1;
}

// ---------------------------------------------------------------- pos_table[cache_positions[k]] = k
__global__ void bbpm_table_kernel(const int* __restrict__ positions,
                                  int* __restrict__ table) {
  const int k = blockIdx.x * blockDim.x + threadIdx.x;
  if (k < KV_CACHE) {
    const int p = positions[NTOK - KV_CACHE + k];
    table[p] = k;
  }
}

// ---------------------------------------------------------------- scatter-add: one wave per token
// lanes 0..31 each handle 4 floats of the 128-float row: coalesced 512B value reads,
// native global_atomic_add_f32 bursts into one L2-resident mem row.
__global__ void bbpm_scatter_kernel(const int* __restrict__ keys,
                                    const float* __restrict__ values,
                                    float* __restrict__ memF) {
  const int t     = blockIdx.x * blockDim.x + threadIdx.x;
  const int token = t >> 5;
  const int chunk = t & 31;
  if (token >= NTOK) return;
  const unsigned slot = ((unsigned)keys[token]) & (BBPM_SIZE - 1);
  const float4 v = ((const float4*)(values + (size_t)token * DIM))[chunk];
  float* dst = memF + (size_t)slot * DIM + (size_t)chunk * 4;
  unsafeAtomicAdd(dst + 0, v.x);
  unsafeAtomicAdd(dst + 1, v.y);
  unsafeAtomicAdd(dst + 2, v.z);
  unsafeAtomicAdd(dst + 3, v.w);
}

// ---------------------------------------------------------------- gather: one wave per output row,
// BATCH rows pipelined through the CDNA5 async global->LDS->global data mover.
// Issue BATCH loads (4 LDS slots) -> one s_wait_asynccnt -> BATCH NT stores -> one wait.
// No VGPR round trip for the payload; reads stay RT (L2-resident mem/cache rows),
// streamed output stores are TH_NT to avoid evicting the read working set from L2.
__global__ void bbpm_gather_kernel(const int* __restrict__ qkeys,
                                   const int* __restrict__ qpos,
                                   const float* __restrict__ values,
                                   const float* __restrict__ memF,
                                   const int* __restrict__ table,
                                   float* __restrict__ out) {
  __shared__ float lds[8 * BATCH * DIM];          // 8 waves x 4 slots x 512B = 16 KB
  const int lane        = threadIdx.x & 31;
  const int wave        = threadIdx.x >> 5;
  const int wavesPerBlk = blockDim.x >> 5;
  const int gwave       = blockIdx.x * wavesPerBlk + wave;
  const int nwaves      = gridDim.x * wavesPerBlk;
  float* slots = &lds[wave * (BATCH * DIM)];

  for (int base = gwave * BATCH; base < NTOK; base += nwaves * BATCH) {
#pragma unroll
    for (int j = 0; j < BATCH; ++j) {
      const int row = base + j;
      if (row < NTOK) {
        const int qp  = qpos[row];
        const int idx = table[qp];
        const float* src;
        if (idx >= 0) {
          src = values + (size_t)(NTOK - KV_CACHE + idx) * DIM;   // v_cache row
        } else {
          const unsigned s = ((unsigned)qkeys[row]) & (BBPM_SIZE - 1);
          src = memF + (size_t)s * DIM;                           // bbpm row (L2 resident)
        }
        gvi4* gsrc = (gvi4*)(uintptr_t)(src + lane * 4);
        lvi4* lp   = (lvi4*)(uint32_t)(uintptr_t)(slots + j * DIM + lane * 4);
        __builtin_amdgcn_global_load_async_to_lds_b128(gsrc, lp, 0, CPOL_RT);
      }
    }
    wait_async0();   // all BATCH loads landed in LDS
#pragma unroll
    for (int j = 0; j < BATCH; ++j) {
      const int row = base + j;
      if (row < NTOK) {
        gvi4* gdst = (gvi4*)(uintptr_t)(out + (size_t)row * DIM + lane * 4);
        lvi4* lp   = (lvi4*)(uint32_t)(uintptr_t)(slots + j * DIM + lane * 4);
        __builtin_amdgcn_global_store_async_from_lds_b128(gdst, lp, 0, CPOL_NT);
      }
    }
    wait_async0();   // stores drained before LDS slots are reused next iteration
  }
}

// ---------------------------------------------------------------- launch
extern "C" void kernel_launch(void* const* d_in, const int* in_sizes, int n_in,
                              void* d_out, int out_size, void* d_ws, size_t ws_size,
                              hipStream_t stream) {
  (void)in_sizes; (void)n_in; (void)out_size; (void)ws_size;

  const int*   keys      = (const int*)d_in[0];
  const float* values    = (const float*)d_in[1];
  const int*   positions = (const int*)d_in[2];
  const int*   qkeys     = (const int*)d_in[3];
  const int*   qpos      = (const int*)d_in[4];
  const float* bbpm      = (const float*)d_in[5];
  float*       out       = (float*)d_out;

  float* memF  = (float*)d_ws;                                        // 64 MB
  int*   table = (int*)((char*)d_ws + (size_t)BBPM_SIZE * DIM * 4);   // +1 MB

  const int threads = 256;

  // 1) mem = bbpm copy, table = -1
  bbpm_init_kernel<<<(BBPM_SIZE * DIM) / threads, threads, 0, stream>>>(bbpm, memF, table);

  // 2) position table for last KV_CACHE tokens
  bbpm_table_kernel<<<(KV_CACHE + threads - 1) / threads, threads, 0, stream>>>(positions, table);

  // 3) scatter-add all token rows into mem (one wave per token)
  bbpm_scatter_kernel<<<(NTOK * 32) / threads, threads, 0, stream>>>(keys, values, memF);

  // 4) gather output rows (async LDS data mover, 4-row pipelined batches)
  bbpm_gather_kernel<<<2048, threads, 0, stream>>>(qkeys, qpos, values, memF, table, out);
}